// POLYGNN_adj_79800492359957
// MI455X (gfx1250) — compile-verified
//
#include <hip/hip_runtime.h>
#include <hip/hip_bf16.h>

// Problem constants (from reference)
#define BB 16
#define MM 128
#define NH 256
#define EH 128
#define HDIM (2 * NH + EH) // 640

typedef __attribute__((ext_vector_type(2))) float v2f;
typedef __attribute__((ext_vector_type(8))) float v8f;

// ---------------------------------------------------------------------------
// Wave-level 16x16 f32 GEMM tile using V_WMMA_F32_16X16X4_F32.
//   C[16,16] = sum_k A[16,K] * B[K,16]
// A addressed as Ab[m*sAm + k*sAk], B as Bb[k*sBk + n] (n stride 1).
// ISA layout (CDNA5 ISA):
//   A 16x4 f32: lanes 0-15 hold M=lane, v0=K0,v1=K1; lanes 16-31 v0=K2,v1=K3
//   B  4x16 f32: row-striped across lanes (symmetric K mapping)
//   C/D: VGPR r -> M = r + 8*(lane>=16), N = lane%16
// ---------------------------------------------------------------------------
__device__ __forceinline__ v8f wave_gemm16(const float* __restrict__ Ab, int sAm, int sAk,
                                           const float* __restrict__ Bb, int sBk,
                                           int K, int lane) {
    const int lr = lane & 15;
    const int hi = lane >> 4;
    const float* Ap = Ab + lr * sAm + (2 * hi) * sAk;
    const float* Bp = Bb + lr + (2 * hi) * sBk;
    v8f c = {};
#pragma unroll 4
    for (int k = 0; k < K; k += 4) {
        v2f av, bv;
        av.x = Ap[0];
        av.y = Ap[sAk];
        bv.x = Bp[0];
        bv.y = Bp[sBk];
        c = __builtin_amdgcn_wmma_f32_16x16x4_f32(false, av, false, bv,
                                                  (short)0, c, false, false);
        Ap += 4 * sAk;
        Bp += 4 * sBk;
    }
    return c;
}

// ---------------------------------------------------------------------------
// k1: aggE[b,j,:] = sum_i A[b,i,j] * E[b,i,j,:] ; sA[b,j] = sum_i A[b,i,j]
// One block per (b,j) column; 4 waves each own i % 4 == wave_id; float4 loads
// (512B per wave per i) stream E exactly once (134 MB -> the roofline cost).
// ---------------------------------------------------------------------------
__global__ void __launch_bounds__(128) k1_agg_edge(const float* __restrict__ E,
                                                   const float* __restrict__ A,
                                                   float* __restrict__ aggE,
                                                   float* __restrict__ sA) {
    const int bj = blockIdx.x;        // 0 .. B*M-1
    const int b = bj / MM;
    const int j = bj % MM;
    const int t = threadIdx.x;
    const int kq = t & 31;            // k-quad: covers k = 4*kq .. 4*kq+3
    const int ig = t >> 5;            // wave id -> i % 4

    const float* __restrict__ Ab = A + (size_t)b * MM * MM + j;                // stride MM over i
    const float* __restrict__ Eb = E + (((size_t)b * MM) * MM + j) * EH + 4 * kq; // stride MM*EH over i

    float4 acc = make_float4(0.f, 0.f, 0.f, 0.f);
    float sa = 0.f;
    for (int i = ig; i < MM; i += 4) {
        const float a = Ab[(size_t)i * MM];
        const float4 e = *(const float4*)(Eb + (size_t)i * MM * EH);
        acc.x += a * e.x;
        acc.y += a * e.y;
        acc.z += a * e.z;
        acc.w += a * e.w;
        sa += a;
    }

    __shared__ float4 red[4][32];
    __shared__ float sred[4];
    red[ig][kq] = acc;
    if (kq == 0) sred[ig] = sa;   // sa identical across lanes of a wave
    __syncthreads();

    if (ig == 0) {
        float4 r0 = red[0][kq], r1 = red[1][kq], r2 = red[2][kq], r3 = red[3][kq];
        float4 r;
        r.x = r0.x + r1.x + r2.x + r3.x;
        r.y = r0.y + r1.y + r2.y + r3.y;
        r.z = r0.z + r1.z + r2.z + r3.z;
        r.w = r0.w + r1.w + r2.w + r3.w;
        *(float4*)(aggE + (size_t)bj * EH + 4 * kq) = r;
        if (t == 0) sA[bj] = sred[0] + sred[1] + sred[2] + sred[3];
    }
}

// ---------------------------------------------------------------------------
// k2: aggX[b] = A[b]^T @ X[b]   ([M,M]^T x [M,NH] per batch), WMMA f32.
// One wave per 16x16 output tile; 2048 tiles total; 512 blocks x 4 waves.
// ---------------------------------------------------------------------------
__global__ void __launch_bounds__(128) k2_aggX(const float* __restrict__ A,
                                               const float* __restrict__ X,
                                               float* __restrict__ aggX) {
    const int lane = threadIdx.x & 31;
    const int wid = threadIdx.x >> 5;
    const int tile = blockIdx.x * 4 + wid;   // 0 .. 2047
    const int b = tile >> 7;                 // / (8*16)
    const int rem = tile & 127;
    const int mt = rem >> 4;                 // j tile 0..7
    const int nt = rem & 15;                 // d tile 0..15

    // A_op[m,k] = A[b, k, mt*16+m]  -> base A + b*M*M + mt*16, sAm=1, sAk=M
    const float* Abase = A + (size_t)b * MM * MM + mt * 16;
    // B_op[k,n] = X[b, k, nt*16+n]  -> base X + b*M*NH + nt*16, sBk=NH
    const float* Bbase = X + (size_t)b * MM * NH + nt * 16;

    v8f c = wave_gemm16(Abase, 1, MM, Bbase, NH, MM, lane);

    const int lr = lane & 15;
    const int hi = lane >> 4;
    float* out = aggX + ((size_t)b * MM + mt * 16) * NH + nt * 16 + lr;
#pragma unroll
    for (int r = 0; r < 8; ++r) {
        const int m = r + 8 * hi;
        out[(size_t)m * NH] = c[r];
    }
}

// ---------------------------------------------------------------------------
// k3: build h[row, 0:640] = relu(concat(agg_node, agg_edge, X)), row = b*M+j
//   tiles [0,2048):       agg_node = aggX @ Wv_w + sA*Wv_b   (K=256, N=256)
//   tiles [2048,3072):    agg_edge = aggE @ We_w + sA*We_b   (K=128, N=128)
//   tiles [3072,5120):    relu copy of X                     (N=256)
// ---------------------------------------------------------------------------
__global__ void __launch_bounds__(128) k3_build_h(const float* __restrict__ aggX,
                                                  const float* __restrict__ aggE,
                                                  const float* __restrict__ sA,
                                                  const float* __restrict__ X,
                                                  const float* __restrict__ Wv_w,
                                                  const float* __restrict__ Wv_b,
                                                  const float* __restrict__ We_w,
                                                  const float* __restrict__ We_b,
                                                  float* __restrict__ h) {
    const int lane = threadIdx.x & 31;
    const int wid = threadIdx.x >> 5;
    const int tile = blockIdx.x * 4 + wid;   // 0 .. 5119
    const int lr = lane & 15;
    const int hi = lane >> 4;

    if (tile < 2048) {
        // agg_node GEMM: rows of aggX (flat 2048) x Wv_w[256,256]
        const int mt = tile >> 4;            // row tile 0..127
        const int nt = tile & 15;            // col tile 0..15
        const int row0 = mt * 16;
        v8f c = wave_gemm16(aggX + (size_t)row0 * NH, NH, 1,
                            Wv_w + nt * 16, NH, NH, lane);
        const int n = nt * 16 + lr;
        const float bias = Wv_b[n];
#pragma unroll
        for (int r = 0; r < 8; ++r) {
            const int row = row0 + r + 8 * hi;
            float v = c[r] + sA[row] * bias;
            h[(size_t)row * HDIM + n] = v > 0.f ? v : 0.f;
        }
    } else if (tile < 3072) {
        // agg_edge GEMM: rows of aggE (flat 2048) x We_w[128,128]
        const int tb = tile - 2048;
        const int mt = tb >> 3;              // row tile 0..127
        const int nt = tb & 7;               // col tile 0..7
        const int row0 = mt * 16;
        v8f c = wave_gemm16(aggE + (size_t)row0 * EH, EH, 1,
                            We_w + nt * 16, EH, EH, lane);
        const int n = nt * 16 + lr;
        const float bias = We_b[n];
#pragma unroll
        for (int r = 0; r < 8; ++r) {
            const int row = row0 + r + 8 * hi;
            float v = c[r] + sA[row] * bias;
            h[(size_t)row * HDIM + NH + n] = v > 0.f ? v : 0.f;
        }
    } else {
        // relu(X) copy into cols [512, 768)
        const int tc = tile - 3072;
        const int mt = tc >> 4;
        const int nt = tc & 15;
        const int n = nt * 16 + lr;
#pragma unroll
        for (int r = 0; r < 8; ++r) {
            const int row = mt * 16 + r + 8 * hi;
            float v = X[(size_t)row * NH + n];
            h[(size_t)row * HDIM + 2 * NH + n] = v > 0.f ? v : 0.f;
        }
    }
}

// ---------------------------------------------------------------------------
// k4: out[row,:] = (h[row,:] @ Wu_w + Wu_b) * w[row]   (K=640, N=256)
// ---------------------------------------------------------------------------
__global__ void __launch_bounds__(128) k4_output(const float* __restrict__ h,
                                                 const float* __restrict__ Wu_w,
                                                 const float* __restrict__ Wu_b,
                                                 const float* __restrict__ w,
                                                 float* __restrict__ out) {
    const int lane = threadIdx.x & 31;
    const int wid = threadIdx.x >> 5;
    const int tile = blockIdx.x * 4 + wid;   // 0 .. 2047
    const int mt = tile >> 4;                // row tile 0..127
    const int nt = tile & 15;                // col tile 0..15
    const int row0 = mt * 16;

    v8f c = wave_gemm16(h + (size_t)row0 * HDIM, HDIM, 1,
                        Wu_w + nt * 16, NH, HDIM, lane);

    const int lr = lane & 15;
    const int hi = lane >> 4;
    const int n = nt * 16 + lr;
    const float bias = Wu_b[n];
#pragma unroll
    for (int r = 0; r < 8; ++r) {
        const int row = row0 + r + 8 * hi;
        out[(size_t)row * NH + n] = (c[r] + bias) * w[row];
    }
}

// ---------------------------------------------------------------------------
// Launcher
// ---------------------------------------------------------------------------
extern "C" void kernel_launch(void* const* d_in, const int* in_sizes, int n_in,
                              void* d_out, int out_size, void* d_ws, size_t ws_size,
                              hipStream_t stream) {
    const float* X    = (const float*)d_in[0];
    const float* E    = (const float*)d_in[1];
    const float* A    = (const float*)d_in[2];
    const float* w    = (const float*)d_in[3];
    const float* Wv_w = (const float*)d_in[4];
    const float* Wv_b = (const float*)d_in[5];
    const float* We_w = (const float*)d_in[6];
    const float* We_b = (const float*)d_in[7];
    const float* Wu_w = (const float*)d_in[8];
    const float* Wu_b = (const float*)d_in[9];
    float* out = (float*)d_out;

    // Workspace layout (floats): aggX | aggE | sA | h   (~8.4 MB total)
    float* ws   = (float*)d_ws;
    float* aggX = ws;                          // B*M*NH = 524288
    float* aggE = aggX + (size_t)BB * MM * NH; // B*M*EH = 262144
    float* sA   = aggE + (size_t)BB * MM * EH; // B*M    = 2048
    float* h    = sA + (size_t)BB * MM;        // B*M*640 = 1310720

    // 1) stream E once: aggE, sA  (bandwidth-bound, the dominant cost)
    k1_agg_edge<<<dim3(BB * MM), dim3(128), 0, stream>>>(E, A, aggE, sA);
    // 2) aggX = A^T @ X per batch (WMMA)
    k2_aggX<<<dim3(512), dim3(128), 0, stream>>>(A, X, aggX);
    // 3) h = relu(concat(aggX@Wv + sA*Wv_b, aggE@We + sA*We_b, X)) (WMMA)
    k3_build_h<<<dim3(1280), dim3(128), 0, stream>>>(aggX, aggE, sA, X,
                                                     Wv_w, Wv_b, We_w, We_b, h);
    // 4) out = (h @ Wu_w + Wu_b) * w (WMMA)
    k4_output<<<dim3(512), dim3(128), 0, stream>>>(h, Wu_w, Wu_b, w, out);
}